// InfFidNet2D_47193100648735
// MI455X (gfx1250) — compile-verified
//
#include <hip/hip_runtime.h>
#include <hip/hip_bf16.h>
#include <math.h>

typedef float v2f __attribute__((ext_vector_type(2)));
typedef float v8f __attribute__((ext_vector_type(8)));

// ---------------------------------------------------------------------------
// fp32 WMMA: D(16x16) += A(16x4) * B(4x16), exact fp32 (matches reference).
// ---------------------------------------------------------------------------
__device__ __forceinline__ v8f wmma4(v2f a, v2f b, v8f c) {
  return __builtin_amdgcn_wmma_f32_16x16x4_f32(
      /*neg_a=*/false, a, /*neg_b=*/false, b,
      /*c_mod=*/(short)0, c, /*reuse_a=*/false, /*reuse_b=*/false);
}

// One wave computes a 16x16 tile of  out = act(in(16xK) @ W(Kact x fan_out) + b)
// s_in: LDS, zero-padded to at least (Kact rounded up to mult of 4) columns.
// W, bias: global, row-major (fan_in x fan_out).
// Main K-loop is unguarded; only the (optional) tail iteration guards W reads.
__device__ __forceinline__ void wave_tile(
    const float* __restrict__ s_in, int ld_in, int Kact,
    const float* __restrict__ W, const float* __restrict__ bias, int fan_out,
    int n0, float* __restrict__ s_out, int ld_out, bool do_tanh)
{
  const int lane = threadIdx.x & 31;
  const int half = lane >> 4;     // lanes 16-31 carry the upper K pair / M+8 rows
  const int mn   = lane & 15;
  v8f c = {0.f, 0.f, 0.f, 0.f, 0.f, 0.f, 0.f, 0.f};

  const int Kmain = Kact & ~3;
  const float* arow = s_in + mn * ld_in + 2 * half;
  const float* wcol = W + (size_t)(2 * half) * fan_out + n0 + mn;
  const size_t wstep = (size_t)4 * fan_out;

  for (int k0 = 0; k0 < Kmain; k0 += 4) {
    v2f a, b;
    a.x = arow[0];
    a.y = arow[1];
    b.x = wcol[0];
    b.y = wcol[fan_out];
    c = wmma4(a, b, c);
    arow += 4;
    wcol += wstep;
  }
  if (Kact & 3) {                       // single guarded tail (Kact in {5,134})
    const int ka = Kmain + 2 * half;
    v2f a, b;
    a.x = arow[0];                      // zero-padded in LDS
    a.y = arow[1];
    b.x = (ka     < Kact) ? wcol[0]       : 0.f;
    b.y = (ka + 1 < Kact) ? wcol[fan_out] : 0.f;
    c = wmma4(a, b, c);
  }

  const float bv = bias[n0 + mn];
#pragma unroll
  for (int v = 0; v < 8; ++v) {
    float val = c[v] + bv;
    if (do_tanh) val = tanhf(val);
    s_out[(v + 8 * half) * ld_out + n0 + mn] = val;
  }
}

// ---------------------------------------------------------------------------
// Kernel 1: h0 = gMLP(bX);  ht = RK4(f, h0) fused per 16-row tile.
// 256 blocks x 512 threads (16 waves). All activations live in LDS.
// ---------------------------------------------------------------------------
#define HT_LDS_FLOATS (2304 + 4096 + 4096 + 4 * 2048 + 128 + 16)

__global__ void __launch_bounds__(512) ht_kernel(
    const float* __restrict__ X, const float* __restrict__ tt,
    const float* gW1, const float* gb1, const float* gW2, const float* gb2,
    const float* gW3, const float* gb3,
    const float* fW1, const float* fb1, const float* fW2, const float* fb2,
    const float* fW3, const float* fb3,
    float* __restrict__ ht)
{
  extern __shared__ float smem[];
  float* s_a   = smem;             // 16 x 144 (f-layer aug input, zero padded)
  float* s_h1  = s_a  + 16 * 144;  // 16 x 256
  float* s_h2  = s_h1 + 16 * 256;  // 16 x 256
  float* s_y0  = s_h2 + 16 * 256;  // 16 x 128
  float* s_yc  = s_y0 + 16 * 128;  // 16 x 128
  float* s_ks  = s_yc + 16 * 128;  // 16 x 128
  float* s_k   = s_ks + 16 * 128;  // 16 x 128
  float* s_bx  = s_k  + 16 * 128;  // 16 x 8 (bX padded)
  float* s_btt = s_bx + 16 * 8;    // 16

  const int tid   = threadIdx.x;
  const int wave  = tid >> 5;
  const int rbase = blockIdx.x * 16;

  if (tid < 16) {
    float t = tt[(rbase + tid) >> 10];   // N = 1024 rows per time step
    if (t == 0.f) t = 1e-8f;
    s_btt[tid] = t;
  }
  for (int i = tid; i < 16 * 8; i += blockDim.x) {
    const int m = i >> 3, k = i & 7;
    s_bx[i] = (k < 5) ? X[(size_t)(rbase + m) * 5 + k] : 0.f;
  }
  for (int i = tid; i < 16 * 144; i += blockDim.x) s_a[i] = 0.f;
  __syncthreads();

  // ---- g-MLP: 5 -> 256 -> 256 -> 128 ----
  wave_tile(s_bx, 8, 5, gW1, gb1, 256, wave * 16, s_h1, 256, true);
  __syncthreads();
  wave_tile(s_h1, 256, 256, gW2, gb2, 256, wave * 16, s_h2, 256, true);
  __syncthreads();
  if (wave < 8)
    wave_tile(s_h2, 256, 256, gW3, gb3, 128, wave * 16, s_y0, 128, false);
  __syncthreads();

  for (int i = tid; i < 16 * 128; i += blockDim.x) {
    s_yc[i] = s_y0[i];
    s_ks[i] = 0.f;
  }
  __syncthreads();

  // ---- one RK4 step of df(t,y) = btt * fMLP([t*btt, y, bX]) ----
  const float stage_t[4] = {0.f, 0.5f, 0.5f, 1.f};
  const float stage_w[4] = {1.f, 2.f, 2.f, 1.f};
  const float stage_c[4] = {0.5f, 0.5f, 1.f, 0.f};

  for (int s = 0; s < 4; ++s) {
    // build aug = [t*btt | y | bX] (134 cols, cols 134..143 stay zero)
    for (int i = tid; i < 16 * 134; i += blockDim.x) {
      const int m = i / 134, k = i % 134;
      float v;
      if (k == 0)        v = stage_t[s] * s_btt[m];
      else if (k < 129)  v = s_yc[m * 128 + (k - 1)];
      else               v = s_bx[m * 8 + (k - 129)];
      s_a[m * 144 + k] = v;
    }
    __syncthreads();
    wave_tile(s_a, 144, 134, fW1, fb1, 256, wave * 16, s_h1, 256, true);
    __syncthreads();
    wave_tile(s_h1, 256, 256, fW2, fb2, 256, wave * 16, s_h2, 256, true);
    __syncthreads();
    if (wave < 8)
      wave_tile(s_h2, 256, 256, fW3, fb3, 128, wave * 16, s_k, 128, false);
    __syncthreads();
    for (int i = tid; i < 16 * 128; i += blockDim.x) {
      const int m = i >> 7;
      const float k = s_btt[m] * s_k[i];
      s_ks[i] += stage_w[s] * k;
      if (s < 3) s_yc[i] = s_y0[i] + stage_c[s] * k;
    }
    __syncthreads();
  }

  for (int i = tid; i < 16 * 128; i += blockDim.x) {
    const int m = i >> 7, n = i & 127;
    ht[(size_t)(rbase + m) * 128 + n] = s_y0[i] + s_ks[i] * (1.f / 6.f);
  }
}

// ---------------------------------------------------------------------------
// Kernel 2: elementwise scalar RK4 on A_mat through the tiny 2->40->40->1 MLP.
// Weights cached in LDS; pure VALU (not matrix shaped per element).
// ---------------------------------------------------------------------------
__device__ __forceinline__ float a_mlp(float c0, float y,
                                       const float* __restrict__ W1,
                                       const float* __restrict__ b1,
                                       const float* __restrict__ W2,
                                       const float* __restrict__ b2,
                                       const float* __restrict__ W3, float b3)
{
  float h1[40];
#pragma unroll 8
  for (int j = 0; j < 40; ++j)
    h1[j] = tanhf(fmaf(c0, W1[j], fmaf(y, W1[40 + j], b1[j])));
  float out = b3;
  for (int j = 0; j < 40; ++j) {
    float acc = b2[j];
#pragma unroll 8
    for (int k = 0; k < 40; ++k) acc = fmaf(h1[k], W2[k * 40 + j], acc);
    out = fmaf(tanhf(acc), W3[j], out);
  }
  return out;
}

#define A_NE 297216   // 129 * 2304

__global__ void __launch_bounds__(256) at_kernel(
    const float* __restrict__ tt, const float* __restrict__ A_mat,
    const float* aW1, const float* ab1, const float* aW2, const float* ab2,
    const float* aW3, const float* ab3, float* __restrict__ At_all)
{
  __shared__ float sW1[80], sb1[40], sW2[1600], sb2[40], sW3[40];
  __shared__ float sb3, stt[4];
  const int tid = threadIdx.x;
  for (int i = tid; i < 80; i += 256) sW1[i] = aW1[i];
  for (int i = tid; i < 40; i += 256) { sb1[i] = ab1[i]; sb2[i] = ab2[i]; sW3[i] = aW3[i]; }
  for (int i = tid; i < 1600; i += 256) sW2[i] = aW2[i];
  if (tid == 0) sb3 = ab3[0];
  if (tid < 4) { float t = tt[tid]; stt[tid] = (t == 0.f) ? 1e-8f : t; }
  __syncthreads();

  const size_t idx = (size_t)blockIdx.x * 256 + tid;
  if (idx >= (size_t)4 * A_NE) return;
  const int    i = (int)(idx / A_NE);
  const size_t j = idx % A_NE;

  const float ta = stt[i];
  const float y0 = A_mat[j];
  const float k1 = ta * a_mlp(0.f,        y0,              sW1, sb1, sW2, sb2, sW3, sb3);
  const float k2 = ta * a_mlp(ta * 0.5f,  y0 + 0.5f * k1,  sW1, sb1, sW2, sb2, sW3, sb3);
  const float k3 = ta * a_mlp(ta * 0.5f,  y0 + 0.5f * k2,  sW1, sb1, sW2, sb2, sW3, sb3);
  const float k4 = ta * a_mlp(ta,         y0 + k3,         sW1, sb1, sW2, sb2, sW3, sb3);
  At_all[idx] = y0 + (k1 + 2.f * k2 + 2.f * k3 + k4) * (1.f / 6.f);
}

// ---------------------------------------------------------------------------
// Kernel 3: pred = ht_i(1024x128) @ Am(128x2304) + Ab  via fp32 WMMA.
// Block = 8 waves, each wave one 16x16 output tile. grid = (18, 64).
// ---------------------------------------------------------------------------
__global__ void __launch_bounds__(256) pred_kernel(
    const float* __restrict__ ht_i, const float* __restrict__ At_i,
    float* __restrict__ pred)
{
  const int lane = threadIdx.x & 31;
  const int wave = threadIdx.x >> 5;                  // 0..7
  const int half = lane >> 4;
  const int mn   = lane & 15;
  const int n0   = (blockIdx.x * 8 + wave) * 16;      // < 2304
  const int m0   = blockIdx.y * 16;                   // < 1024
  v8f c = {0.f, 0.f, 0.f, 0.f, 0.f, 0.f, 0.f, 0.f};

  const float* arow = ht_i + (size_t)(m0 + mn) * 128 + 2 * half;
  const float* wcol = At_i + (size_t)(2 * half) * 2304 + n0 + mn;
#pragma unroll 4
  for (int k0 = 0; k0 < 128; k0 += 4) {
    v2f a, b;
    a.x = arow[0];
    a.y = arow[1];
    b.x = wcol[0];
    b.y = wcol[2304];
    c = wmma4(a, b, c);
    arow += 4;
    wcol += (size_t)4 * 2304;
  }
  const float bv = At_i[(size_t)128 * 2304 + n0 + mn];   // bias row Ab
#pragma unroll
  for (int v = 0; v < 8; ++v)
    pred[(size_t)(m0 + v + 8 * half) * 2304 + n0 + mn] = c[v] + bv;
}

// ---------------------------------------------------------------------------
// Kernel 4: bilinear resize 48x48 -> fid x fid (half-pixel, clamped), per image.
// ---------------------------------------------------------------------------
__global__ void __launch_bounds__(256) resize_kernel(
    const float* __restrict__ pred, float* __restrict__ out, int fid)
{
  const int n = blockIdx.y;
  const int p = blockIdx.x * 256 + threadIdx.x;
  const int tot = fid * fid;
  if (p >= tot) return;
  const int r = p / fid, c = p % fid;
  const float scale = 48.f / (float)fid;
  const float sr = (r + 0.5f) * scale - 0.5f;
  const float sc = (c + 0.5f) * scale - 0.5f;
  const float fr0 = floorf(sr), fc0 = floorf(sc);
  const int r0 = (int)fr0, c0 = (int)fc0;
  const float wr = sr - fr0, wc = sc - fc0;
  const int r0c = min(max(r0, 0), 47),     r1c = min(max(r0 + 1, 0), 47);
  const int c0c = min(max(c0, 0), 47),     c1c = min(max(c0 + 1, 0), 47);
  const float* P = pred + (size_t)n * 2304;
  const float v00 = P[r0c * 48 + c0c], v01 = P[r0c * 48 + c1c];
  const float v10 = P[r1c * 48 + c0c], v11 = P[r1c * 48 + c1c];
  const float v0 = v00 + wc * (v01 - v00);
  const float v1 = v10 + wc * (v11 - v10);
  out[(size_t)n * tot + p] = v0 + wr * (v1 - v0);
}

// ---------------------------------------------------------------------------
// Launch. Inputs (pytree order):
//  0:X 1:tt 2-7:g(W1,b1,W2,b2,W3,b3) 8-13:f(...) 14-19:A(...) 20:A_mat
// ---------------------------------------------------------------------------
extern "C" void kernel_launch(void* const* d_in, const int* in_sizes, int n_in,
                              void* d_out, int out_size, void* d_ws, size_t ws_size,
                              hipStream_t stream) {
  (void)in_sizes; (void)n_in; (void)out_size; (void)ws_size;
  const float* X   = (const float*)d_in[0];
  const float* tt  = (const float*)d_in[1];
  const float* gW1 = (const float*)d_in[2];  const float* gb1 = (const float*)d_in[3];
  const float* gW2 = (const float*)d_in[4];  const float* gb2 = (const float*)d_in[5];
  const float* gW3 = (const float*)d_in[6];  const float* gb3 = (const float*)d_in[7];
  const float* fW1 = (const float*)d_in[8];  const float* fb1 = (const float*)d_in[9];
  const float* fW2 = (const float*)d_in[10]; const float* fb2 = (const float*)d_in[11];
  const float* fW3 = (const float*)d_in[12]; const float* fb3 = (const float*)d_in[13];
  const float* aW1 = (const float*)d_in[14]; const float* ab1 = (const float*)d_in[15];
  const float* aW2 = (const float*)d_in[16]; const float* ab2 = (const float*)d_in[17];
  const float* aW3 = (const float*)d_in[18]; const float* ab3 = (const float*)d_in[19];
  const float* A_mat = (const float*)d_in[20];

  float* ws   = (float*)d_ws;
  float* ht   = ws;                                   // 4096*128
  float* At   = ws + (size_t)4096 * 128;              // 4*297216
  float* pred = At + (size_t)4 * A_NE;                // 1024*2304 (reused per i)
  float* out  = (float*)d_out;

  ht_kernel<<<dim3(256), dim3(512), HT_LDS_FLOATS * sizeof(float), stream>>>(
      X, tt, gW1, gb1, gW2, gb2, gW3, gb3, fW1, fb1, fW2, fb2, fW3, fb3, ht);

  at_kernel<<<dim3((4 * A_NE + 255) / 256), dim3(256), 0, stream>>>(
      tt, A_mat, aW1, ab1, aW2, ab2, aW3, ab3, At);

  const int fids[4] = {24, 32, 40, 48};
  size_t ooff = 0;
  for (int i = 0; i < 4; ++i) {
    const float* ht_i = ht + (size_t)i * 1024 * 128;
    const float* At_i = At + (size_t)i * A_NE;
    pred_kernel<<<dim3(18, 64), dim3(256), 0, stream>>>(ht_i, At_i, pred);
    const int fid = fids[i];
    resize_kernel<<<dim3((fid * fid + 255) / 256, 1024), dim3(256), 0, stream>>>(
        pred, out + ooff, fid);
    ooff += (size_t)1024 * fid * fid;
  }
}